// Sinkhorn_sparse_rpcl_39573828665617
// MI455X (gfx1250) — compile-verified
//
#include <hip/hip_runtime.h>
#include <hip/hip_bf16.h>
#include <stddef.h>

#define RR 2048
#define CC 8192
#define EPS_F 50.0f
#define NEGV -1e30f
#define NROUNDS 24

typedef __attribute__((ext_vector_type(2))) float v2f;
typedef __attribute__((ext_vector_type(8))) float v8f;

// ---------------- elementwise kernels ----------------

__global__ void k_exp(const float* __restrict__ in, float* __restrict__ s, int n4) {
  int i = blockIdx.x * blockDim.x + threadIdx.x;
  if (i >= n4) return;
  float4 x = reinterpret_cast<const float4*>(in)[i];
  float4 y;
  y.x = __expf(x.x * EPS_F);
  y.y = __expf(x.y * EPS_F);
  y.z = __expf(x.z * EPS_F);
  y.w = __expf(x.w * EPS_F);
  reinterpret_cast<float4*>(s)[i] = y;
}

__global__ void k_fill_f32(float* __restrict__ p, int n, float v) {
  int i = blockIdx.x * blockDim.x + threadIdx.x;
  if (i < n) p[i] = v;
}

__global__ void k_fill_i32(int* __restrict__ p, int n, int v) {
  int i = blockIdx.x * blockDim.x + threadIdx.x;
  if (i < n) p[i] = v;
}

// ---------------- row sums via V_WMMA_F32_16X16X4_F32 ----------------
// D = A(16x4) * ones(4x16) + C  =>  D[m][n] = C + sum_k A[m][k]  (row sums).
// A layout (ISA 7.12.2, 32-bit A 16x4): lanes 0-15 = rows, lanes 16-31 = rows
// again with K offset 2; VGPR0/VGPR1 hold consecutive K. So lane l loads the
// two floats at column (k + 2*(l>>4)) of row (l&15).
// D layout (32-bit C/D 16x16): VGPR j, lanes 0-15 -> row j; lanes 16-31 -> row j+8.
// grid: RR/16 blocks of 256 threads (8 waves); wave w covers K-slab of CC/8.
__global__ void k_rowsum_wmma(const float* __restrict__ s, float* __restrict__ rowsum) {
  int wave = threadIdx.x >> 5;
  int lane = threadIdx.x & 31;
  int rbase = blockIdx.x * 16;
  int kslab = CC / 8;
  int k0 = wave * kslab;
  int m = lane & 15;
  int koff = (lane >> 4) * 2;
  const float* rowp = s + (size_t)(rbase + m) * CC + koff;
  v8f acc = {};
  v2f b;
  b.x = 1.0f; b.y = 1.0f;
  for (int k = k0; k < k0 + kslab; k += 4) {
    v2f a;
    a.x = rowp[k];
    a.y = rowp[k + 1];
    acc = __builtin_amdgcn_wmma_f32_16x16x4_f32(
        /*neg_a=*/false, a, /*neg_b=*/false, b,
        /*c_mod=*/(short)0, acc, /*reuse_a=*/false, /*reuse_b=*/false);
  }
  if (lane == 0) {
    for (int j = 0; j < 8; ++j) atomicAdd(&rowsum[rbase + j], acc[j]);
  } else if (lane == 16) {
    for (int j = 0; j < 8; ++j) atomicAdd(&rowsum[rbase + 8 + j], acc[j]);
  }
}

__global__ void k_rowscale(float* __restrict__ s, const float* __restrict__ rowsum, int n4) {
  int i = blockIdx.x * blockDim.x + threadIdx.x;
  if (i >= n4) return;
  int row = (i * 4) >> 13;           // CC == 8192 == 2^13
  float inv = 1.0f / rowsum[row];
  float4 v = reinterpret_cast<float4*>(s)[i];
  v.x *= inv; v.y *= inv; v.z *= inv; v.w *= inv;
  reinterpret_cast<float4*>(s)[i] = v;
}

// ---------------- column sums (coalesced VALU) ----------------
// grid: (CC/256, 8) ; block 256. Thread owns one column, sums a 256-row slab.
__global__ void k_colsum(const float* __restrict__ s, float* __restrict__ colsum) {
  int c = blockIdx.x * blockDim.x + threadIdx.x;
  int r0 = blockIdx.y * (RR / 8);
  float acc = 0.0f;
  const float* p = s + (size_t)r0 * CC + c;
  for (int r = 0; r < RR / 8; ++r) acc += p[(size_t)r * CC];
  atomicAdd(&colsum[c], acc);
}

__global__ void k_colscale(float* __restrict__ s, const float* __restrict__ colsum, int n4) {
  int i = blockIdx.x * blockDim.x + threadIdx.x;
  if (i >= n4) return;
  int cb = (i * 4) & (CC - 1);
  float4 cs = *reinterpret_cast<const float4*>(&colsum[cb]);
  float4 v = reinterpret_cast<float4*>(s)[i];
  v.x /= cs.x; v.y /= cs.y; v.z /= cs.z; v.w /= cs.w;
  reinterpret_cast<float4*>(s)[i] = v;
}

// ---------------- top-10 per row ----------------
__global__ void k_top10(const float* __restrict__ s, float* __restrict__ outc) {
  int row = blockIdx.x;
  int tid = threadIdx.x;
  const float* rp = s + (size_t)row * CC;
  __shared__ float sv[256];
  __shared__ int si[256];
  __shared__ int chosen[10];
  for (int p = 0; p < 10; ++p) {
    float bv = NEGV;
    int bi = CC;
    for (int c = tid; c < CC; c += 256) {
      bool skip = false;
      for (int q = 0; q < p; ++q)
        if (chosen[q] == c) skip = true;
      if (!skip) {
        float v = rp[c];
        if (v > bv || (v == bv && c < bi)) { bv = v; bi = c; }
      }
    }
    sv[tid] = bv; si[tid] = bi;
    __syncthreads();
    for (int off = 128; off > 0; off >>= 1) {
      if (tid < off) {
        float ov = sv[tid + off]; int oi = si[tid + off];
        if (ov > sv[tid] || (ov == sv[tid] && oi < si[tid])) { sv[tid] = ov; si[tid] = oi; }
      }
      __syncthreads();
    }
    if (tid == 0) {
      chosen[p] = si[0];
      outc[row * 10 + p] = (float)si[0];
    }
    __syncthreads();
  }
}

// ---------------- greedy mutual-argmax matching ----------------

// block per row: best unused column for each unmatched row
__global__ void k_rowbest(const float* __restrict__ s, const int* __restrict__ col,
                          const int* __restrict__ used, int* __restrict__ c1) {
  int row = blockIdx.x;
  int tid = threadIdx.x;
  if (col[row] >= 0) {                 // uniform branch across the block
    if (tid == 0) c1[row] = -1;
    return;
  }
  const float* rp = s + (size_t)row * CC;
  float bv = NEGV;
  int bi = CC;
  for (int c = tid; c < CC; c += 256) {
    if (!used[c]) {
      float v = rp[c];
      if (v > bv || (v == bv && c < bi)) { bv = v; bi = c; }
    }
  }
  __shared__ float sv[256];
  __shared__ int si[256];
  sv[tid] = bv; si[tid] = bi;
  __syncthreads();
  for (int off = 128; off > 0; off >>= 1) {
    if (tid < off) {
      float ov = sv[tid + off]; int oi = si[tid + off];
      if (ov > sv[tid] || (ov == sv[tid] && oi < si[tid])) { sv[tid] = ov; si[tid] = oi; }
    }
    __syncthreads();
  }
  if (tid == 0) c1[row] = (si[0] < CC) ? si[0] : -1;
}

// thread per column: best unmatched row for each column (coalesced across lanes)
__global__ void k_colbest(const float* __restrict__ s, const int* __restrict__ col,
                          int* __restrict__ rbi) {
  int c = blockIdx.x * blockDim.x + threadIdx.x;
  float bv = NEGV;
  int bi = -1;
  for (int r = 0; r < RR; ++r) {
    if (col[r] < 0) {
      float v = s[(size_t)r * CC + c];
      if (v > bv) { bv = v; bi = r; }   // ties keep first (lowest r)
    }
  }
  rbi[c] = bi;
}

// claim mutual pairs: row r and column c pick each other
__global__ void k_mutual(int* __restrict__ col, int* __restrict__ used,
                         const int* __restrict__ c1, const int* __restrict__ rbi) {
  int r = blockIdx.x * blockDim.x + threadIdx.x;
  if (r >= RR) return;
  if (col[r] < 0) {
    int c = c1[r];
    if (c >= 0 && rbi[c] == r) {
      col[r] = c;        // c unique to r (rbi[c]==r), no write race
      used[c] = 1;
    }
  }
}

// single-block serial cleanup for any rows still unmatched
__global__ void k_cleanup(const float* __restrict__ s, int* __restrict__ col,
                          int* __restrict__ used) {
  int tid = threadIdx.x;
  __shared__ float sv[256];
  __shared__ int si[256];
  for (int r = 0; r < RR; ++r) {
    if (col[r] >= 0) continue;         // uniform across block
    const float* rp = s + (size_t)r * CC;
    float bv = NEGV;
    int bi = CC;
    for (int c = tid; c < CC; c += 256) {
      if (!used[c]) {
        float v = rp[c];
        if (v > bv || (v == bv && c < bi)) { bv = v; bi = c; }
      }
    }
    sv[tid] = bv; si[tid] = bi;
    __syncthreads();
    for (int off = 128; off > 0; off >>= 1) {
      if (tid < off) {
        float ov = sv[tid + off]; int oi = si[tid + off];
        if (ov > sv[tid] || (ov == sv[tid] && oi < si[tid])) { sv[tid] = ov; si[tid] = oi; }
      }
      __syncthreads();
    }
    if (tid == 0 && si[0] < CC) {
      col[r] = si[0];
      used[si[0]] = 1;
    }
    __syncthreads();
  }
}

// ---------------- output assembly ----------------

__global__ void k_zero_out(float* __restrict__ a, int n4) {
  int i = blockIdx.x * blockDim.x + threadIdx.x;
  if (i >= n4) return;
  float4 z = {0.f, 0.f, 0.f, 0.f};
  reinterpret_cast<float4*>(a)[i] = z;
}

__global__ void k_scatter(const int* __restrict__ col, float* __restrict__ out) {
  int r = blockIdx.x * blockDim.x + threadIdx.x;
  if (r >= RR) return;
  int c = col[r];
  if (c >= 0 && c < CC) out[(size_t)r * CC + c] = 1.0f;
}

// ---------------- launcher ----------------

extern "C" void kernel_launch(void* const* d_in, const int* in_sizes, int n_in,
                              void* d_out, int out_size, void* d_ws, size_t ws_size,
                              hipStream_t stream) {
  (void)in_sizes; (void)n_in; (void)out_size; (void)ws_size;
  const float* sims = (const float*)d_in[0];
  float* out = (float*)d_out;

  // Stage s (2048x8192 f32 = 64 MiB, L2-resident) directly in the assign
  // region of d_out; it is zeroed+scattered at the end after all readers.
  float* s = out;
  float* out_cols = out + (size_t)RR * CC;

  // Small workspace slices in d_ws (~120 KB total).
  float* rowsum = (float*)d_ws;            // RR f32
  float* colsum = rowsum + RR;             // CC f32
  int* colarr   = (int*)(colsum + CC);     // RR i32 : row -> matched column
  int* used     = colarr + RR;             // CC i32 : column taken
  int* c1       = used + CC;               // RR i32 : row's current best column
  int* rbi      = c1 + RR;                 // CC i32 : column's current best row

  const int n4 = (RR * CC) / 4;
  const int TPB = 256;

  // 1) s = exp(50 * sims)
  k_exp<<<n4 / TPB, TPB, 0, stream>>>(sims, s, n4);

  // 2) 10 Sinkhorn iterations: row-norm (even), col-norm (odd)
  for (int it = 0; it < 10; ++it) {
    if ((it & 1) == 0) {
      k_fill_f32<<<RR / TPB, TPB, 0, stream>>>(rowsum, RR, 0.0f);
      k_rowsum_wmma<<<RR / 16, TPB, 0, stream>>>(s, rowsum);
      k_rowscale<<<n4 / TPB, TPB, 0, stream>>>(s, rowsum, n4);
    } else {
      k_fill_f32<<<CC / TPB, TPB, 0, stream>>>(colsum, CC, 0.0f);
      k_colsum<<<dim3(CC / TPB, 8), TPB, 0, stream>>>(s, colsum);
      k_colscale<<<n4 / TPB, TPB, 0, stream>>>(s, colsum, n4);
    }
  }

  // 3) top-10 columns per row
  k_top10<<<RR, TPB, 0, stream>>>(s, out_cols);

  // 4) greedy duplicate-free matching (mutual-argmax rounds + serial cleanup)
  k_fill_i32<<<RR / TPB, TPB, 0, stream>>>(colarr, RR, -1);
  k_fill_i32<<<CC / TPB, TPB, 0, stream>>>(used, CC, 0);
  for (int round = 0; round < NROUNDS; ++round) {
    k_rowbest<<<RR, TPB, 0, stream>>>(s, colarr, used, c1);
    k_colbest<<<CC / TPB, TPB, 0, stream>>>(s, colarr, rbi);
    k_mutual<<<RR / TPB, TPB, 0, stream>>>(colarr, used, c1, rbi);
  }
  k_cleanup<<<1, TPB, 0, stream>>>(s, colarr, used);

  // 5) overwrite staging region with the one-hot assign matrix
  k_zero_out<<<n4 / TPB, TPB, 0, stream>>>(out, n4);
  k_scatter<<<RR / TPB, TPB, 0, stream>>>(colarr, out);
}